// GraphEncoding_6880537608484
// MI455X (gfx1250) — compile-verified
//
#include <hip/hip_runtime.h>
#include <hip/hip_bf16.h>

// ---------------------------------------------------------------------------
// CDNA5 (gfx1250) GCN encoder.
//  - Dense X@W via V_WMMA_F32_16X16X4_F32 (full fp32: the 10 GEMMs total
//    ~10 GFLOP, microseconds at any WMMA rate; the workload is bound by the
//    1.6M-edge gather/scatter-add (~7 GB random traffic @ 23.3 TB/s HBM,
//    mostly L2-resident since feature buffers are 51 MB << 192 MB L2),
//    so fp32 exactness is free)
//  - Graph aggregation via native fp32 global atomics (edge-parallel scatter)
//  - relu / addend / Cin are template parameters: no selects in the K-loop,
//    full unroll -> straight-line load + v_wmma chains.
// ---------------------------------------------------------------------------

typedef float v2f __attribute__((ext_vector_type(2)));
typedef float v8f __attribute__((ext_vector_type(8)));

// ---------------------------------------------------------------------------
// GEMM: OUT[N,Cout] = (RELU ? relu(X) : X)[N,CIN] @ W[CIN,Cout] (+ addend)
// One wave -> one 16x16 output tile.  K-loop in steps of 4, fully unrolled.
// A layout (16x4 f32, ISA 7.12.2): lanes 0-15 hold K=0,1; lanes 16-31 K=2,3.
// B layout mirrors A with N across lanes.  C/D: VGPR v = row v / v+8.
// Requires N % 16 == 0, CIN % 4 == 0, Cout % 16 == 0 (true for this model).
// ---------------------------------------------------------------------------
template <int CIN, bool RELU, bool HAS_ADD>
__global__ void __launch_bounds__(128)
gemm_wmma_f32(const float* __restrict__ X, const float* __restrict__ W,
              const float* __restrict__ addend, float* __restrict__ OUT,
              int N, int Cout)
{
    const int lane    = threadIdx.x & 31;
    const int wave    = threadIdx.x >> 5;
    const int rowTile = blockIdx.x * 4 + wave;
    const int nRowTiles = N >> 4;
    if (rowTile >= nRowTiles) return;          // wave-uniform: EXEC stays all-1s

    const int l16 = lane & 15;
    const int hi  = lane >> 4;                 // 0: K=0,1   1: K=2,3
    const int row = (rowTile << 4) + l16;      // A row held by this lane
    const int col = (blockIdx.y << 4) + l16;   // B/C/D column held by this lane

    const float* __restrict__ Xrow = X + (size_t)row * CIN;
    const float* __restrict__ Wc   = W + col;

    v8f acc = {};
#pragma unroll
    for (int k0 = 0; k0 < CIN; k0 += 4) {
        const int ka = k0 + (hi << 1);
        v2f a = *(const v2f*)(Xrow + ka);
        if (RELU) { a.x = fmaxf(a.x, 0.0f); a.y = fmaxf(a.y, 0.0f); }
        v2f b;
        b.x = Wc[(size_t)ka * Cout];
        b.y = Wc[(size_t)(ka + 1) * Cout];
        // (neg_a, A, neg_b, B, c_mod, C, reuse_a, reuse_b)
        acc = __builtin_amdgcn_wmma_f32_16x16x4_f32(
                  false, a, false, b, (short)0, acc, false, false);
    }

    const int mBase = (rowTile << 4) + (hi << 3);  // rows v..v+7 of this half
#pragma unroll
    for (int v = 0; v < 8; ++v) {
        const size_t idx = (size_t)(mBase + v) * Cout + col;
        OUT[idx] = HAS_ADD ? (acc[v] + addend[idx]) : acc[v];
    }
}

// ---------------------------------------------------------------------------
// Small helper kernels
// ---------------------------------------------------------------------------
__global__ void fill_f32(float* __restrict__ p, float v, int n)
{
    int i = blockIdx.x * blockDim.x + threadIdx.x;
    if (i < n) p[i] = v;
}

__global__ void deg_scatter(const int* __restrict__ dst, float* __restrict__ deg, int E)
{
    int i = blockIdx.x * blockDim.x + threadIdx.x;
    if (i < E) unsafeAtomicAdd(&deg[dst[i]], 1.0f);
}

__global__ void finalize_dinv(float* __restrict__ deg, int N)
{
    int i = blockIdx.x * blockDim.x + threadIdx.x;
    if (i < N) deg[i] = rsqrtf(fmaxf(deg[i], 1.0f));
}

__global__ void compute_norm(const int* __restrict__ src, const int* __restrict__ dst,
                             const float* __restrict__ dinv, float* __restrict__ norm, int E)
{
    int i = blockIdx.x * blockDim.x + threadIdx.x;
    if (i < E) norm[i] = dinv[src[i]] * dinv[dst[i]];
}

// OUT[n][c] = bias[c] + T[n][c] * dinv[n]^2      (bias + self-loop term, no atomics)
__global__ void seed_bias_selfloop(const float* __restrict__ T, const float* __restrict__ bias,
                                   const float* __restrict__ dinv, float* __restrict__ OUT,
                                   int total, int lgC, int C)
{
    int i = blockIdx.x * blockDim.x + threadIdx.x;
    if (i >= total) return;
    const int n = i >> lgC;
    const int c = i & (C - 1);
    const float di = dinv[n];
    OUT[i] = bias[c] + T[i] * di * di;
}

// Edge-parallel scatter-add: OUT[dst[e]][c..c+3] += T[src[e]][c..c+3] * norm[e]
__global__ void edge_scatter(const float* __restrict__ T, const int* __restrict__ src,
                             const int* __restrict__ dst, const float* __restrict__ norm,
                             float* __restrict__ OUT, long long total, int lgC4, int C)
{
    long long tid = (long long)blockIdx.x * blockDim.x + threadIdx.x;
    if (tid >= total) return;
    const int e  = (int)(tid >> lgC4);
    const int cg = ((int)tid & ((1 << lgC4) - 1)) << 2;
    const int s = src[e];
    const int d = dst[e];
    const float w = norm[e];
    const float4 v = *(const float4*)(T + (size_t)s * C + cg);
    float* o = OUT + (size_t)d * C + cg;
    unsafeAtomicAdd(o + 0, v.x * w);
    unsafeAtomicAdd(o + 1, v.y * w);
    unsafeAtomicAdd(o + 2, v.z * w);
    unsafeAtomicAdd(o + 3, v.w * w);
}

__global__ void add_inplace(float* __restrict__ a, const float* __restrict__ b, int n)
{
    int i = blockIdx.x * blockDim.x + threadIdx.x;
    if (i < n) a[i] += b[i];
}

// ---------------------------------------------------------------------------
// Host orchestration
// ---------------------------------------------------------------------------
extern "C" void kernel_launch(void* const* d_in, const int* in_sizes, int n_in,
                              void* d_out, int out_size, void* d_ws, size_t ws_size,
                              hipStream_t stream)
{
    const float* x     = (const float*)d_in[0];
    const int*   ei    = (const int*)  d_in[1];
    const float* W_in  = (const float*)d_in[2];
    const float* b_in  = (const float*)d_in[3];
    const float* W0a   = (const float*)d_in[4];
    const float* b0a   = (const float*)d_in[5];
    const float* W0b   = (const float*)d_in[6];
    const float* b0b   = (const float*)d_in[7];
    const float* W1a   = (const float*)d_in[8];
    const float* b1a   = (const float*)d_in[9];
    const float* W1b   = (const float*)d_in[10];
    const float* b1b   = (const float*)d_in[11];
    const float* Wsk1  = (const float*)d_in[12];
    const float* W2a   = (const float*)d_in[13];
    const float* b2a   = (const float*)d_in[14];
    const float* W2b   = (const float*)d_in[15];
    const float* b2b   = (const float*)d_in[16];
    const float* Wsk2  = (const float*)d_in[17];
    const float* W_out = (const float*)d_in[18];
    const float* b_out = (const float*)d_in[19];
    float* out = (float*)d_out;

    const int N = in_sizes[0] / 64;   // 100000 (divisible by 16)
    const int E = in_sizes[1] / 2;    // 1600000
    const int* src = ei;              // edge_index[0]
    const int* dst = ei + E;          // edge_index[1]

    // Workspace layout (floats)
    float* ws   = (float*)d_ws;
    float* dinv = ws;                          // N      (deg -> dinv in place)
    float* norm = dinv + N;                    // E
    float* bufA = norm + E;                    // N*128
    float* bufB = bufA + (size_t)N * 128;      // N*128
    float* bufC = bufB + (size_t)N * 128;      // N*128
    float* T    = bufC + (size_t)N * 128;      // N*128 (pre-aggregation X@W)

    const int TB = 256;
    auto nblk = [](long long n, int tb) { return (unsigned)((n + tb - 1) / tb); };

    // --- graph normalization (computed once, reused by all 8 layers) ---
    fill_f32<<<nblk(N, TB), TB, 0, stream>>>(dinv, 1.0f, N);          // self-loop
    deg_scatter<<<nblk(E, TB), TB, 0, stream>>>(dst, dinv, E);
    finalize_dinv<<<nblk(N, TB), TB, 0, stream>>>(dinv, N);
    compute_norm<<<nblk(E, TB), TB, 0, stream>>>(src, dst, dinv, norm, E);

    const int nRowTiles = N / 16;

    // Static dispatch over (Cin, relu, addend) -> fully unrolled WMMA kernels
    auto gemm = [&](const float* X, const float* W, const float* addend, float* O,
                    int Cin, int Cout, int relu) {
        dim3 grid((nRowTiles + 3) / 4, Cout / 16);
        #define GEMM_CASE(CINV)                                                        \
            if (Cin == CINV) {                                                         \
                if (addend)                                                            \
                    gemm_wmma_f32<CINV, false, true><<<grid, 128, 0, stream>>>(        \
                        X, W, addend, O, N, Cout);                                     \
                else if (relu)                                                         \
                    gemm_wmma_f32<CINV, true, false><<<grid, 128, 0, stream>>>(        \
                        X, W, nullptr, O, N, Cout);                                    \
                else                                                                   \
                    gemm_wmma_f32<CINV, false, false><<<grid, 128, 0, stream>>>(       \
                        X, W, nullptr, O, N, Cout);                                    \
                return;                                                                \
            }
        GEMM_CASE(32)
        GEMM_CASE(64)
        GEMM_CASE(128)
        #undef GEMM_CASE
    };

    // One GCNConv: O = scatter(norm, (relu? relu(X):X) @ W) + b, self-loops folded in
    auto gcn = [&](const float* X, const float* W, const float* b, float* O,
                   int Cin, int Cout, int relu) {
        gemm(X, W, nullptr, T, Cin, Cout, relu);
        const int lgC = (Cout == 32) ? 5 : (Cout == 64) ? 6 : 7;
        const int nc = N * Cout;
        seed_bias_selfloop<<<nblk(nc, TB), TB, 0, stream>>>(T, b, dinv, O, nc, lgC, Cout);
        const long long tot = (long long)E << (lgC - 2);
        edge_scatter<<<nblk(tot, TB), TB, 0, stream>>>(T, src, dst, norm, O, tot, lgC - 2, Cout);
    };

    // conv_in: 64 -> 32
    gcn(x, W_in, b_in, bufA, 64, 32, 0);                 // h   = bufA [N,32]

    // level 0: 32 -> 32, identity shortcut
    gcn(bufA, W0a, b0a, bufB, 32, 32, 0);                // g   = bufB
    gcn(bufB, W0b, b0b, bufC, 32, 32, 1);                // r   = bufC
    add_inplace<<<nblk(N * 32, TB), TB, 0, stream>>>(bufA, bufC, N * 32);  // h += r

    // level 1: 32 -> 64, projection shortcut
    gcn(bufA, W1a, b1a, bufB, 32, 64, 0);                // g   = bufB [N,64]
    gcn(bufB, W1b, b1b, bufC, 64, 64, 1);                // r   = bufC [N,64]
    gemm(bufA, Wsk1, bufC, bufB, 32, 64, 0);             // h64 = h@Wsk1 + r -> bufB

    // level 2: 64 -> 128, projection shortcut
    gcn(bufB, W2a, b2a, bufA, 64, 128, 0);               // g    = bufA [N,128]
    gcn(bufA, W2b, b2b, bufC, 128, 128, 1);              // r    = bufC [N,128]
    gemm(bufB, Wsk2, bufC, bufA, 64, 128, 0);            // h128 = h64@Wsk2 + r -> bufA

    // conv_out: 128 -> 32 with relu on input
    gcn(bufA, W_out, b_out, out, 128, 32, 1);
}